// TranslationRotationLoss_3186865733865
// MI455X (gfx1250) — compile-verified
//
#include <hip/hip_runtime.h>
#include <hip/hip_bf16.h>
#include <math.h>

typedef __attribute__((ext_vector_type(2))) float v2f;
typedef __attribute__((ext_vector_type(8))) float v8f;

#define BLOCK 256
#define ROWS_PER_THREAD 4

// Per-row loss contribution, pre-scaled so the global sum is the final answer.
//   translation: sum of squared diffs over 3 comps, weighted by 1/(3B)
//   rotation:    (2*acos(clip(|<qp,qt>|/(|qp||qt|),0,1)))^2, weighted by 1/B
__device__ __forceinline__ float row_contrib(const float* __restrict__ p,
                                             const float* __restrict__ t,
                                             float invB3, float invB) {
    float d0 = p[0] - t[0];
    float d1 = p[1] - t[1];
    float d2 = p[2] - t[2];
    float tl = d0 * d0 + d1 * d1 + d2 * d2;

    float dot = p[3] * t[3] + p[4] * t[4] + p[5] * t[5] + p[6] * t[6];
    float np  = p[3] * p[3] + p[4] * p[4] + p[5] * p[5] + p[6] * p[6];
    float nt  = t[3] * t[3] + t[4] * t[4] + t[5] * t[5] + t[6] * t[6];
    float c   = fabsf(dot) * rsqrtf(np * nt);   // >= 0 by construction
    c = fminf(c, 1.0f);                         // clip for rsqrt overshoot
    float ang = 2.0f * acosf(c);
    return tl * invB3 + ang * ang * invB;
}

__global__ void __launch_bounds__(BLOCK)
pose_loss_main(const float* __restrict__ pred, const float* __restrict__ tgt,
               float* __restrict__ partials, long nrows, float invB3, float invB) {
    const long tid      = (long)blockIdx.x * BLOCK + threadIdx.x;
    const long base_row = tid * ROWS_PER_THREAD;

    float local = 0.0f;

    if (base_row + ROWS_PER_THREAD <= nrows) {
        // 4 rows = 28 floats = 7 x float4, 16B-aligned (tid*112 bytes).
        const float4* p4 = reinterpret_cast<const float4*>(pred + base_row * 7);
        const float4* t4 = reinterpret_cast<const float4*>(tgt  + base_row * 7);
        float pb[28], tb[28];
#pragma unroll
        for (int i = 0; i < 7; ++i) {
            float4 v = p4[i];
            pb[4 * i + 0] = v.x; pb[4 * i + 1] = v.y;
            pb[4 * i + 2] = v.z; pb[4 * i + 3] = v.w;
        }
#pragma unroll
        for (int i = 0; i < 7; ++i) {
            float4 v = t4[i];
            tb[4 * i + 0] = v.x; tb[4 * i + 1] = v.y;
            tb[4 * i + 2] = v.z; tb[4 * i + 3] = v.w;
        }
#pragma unroll
        for (int r = 0; r < ROWS_PER_THREAD; ++r)
            local += row_contrib(pb + 7 * r, tb + 7 * r, invB3, invB);
    } else if (base_row < nrows) {
        // Generic tail (never taken for B = 8388608, kept for correctness).
        for (long r = base_row; r < nrows; ++r)
            local += row_contrib(pred + r * 7, tgt + r * 7, invB3, invB);
    }

    // ---- Wave32 reduction via V_WMMA_F32_16X16X4_F32 ----
    // A (16x4 f32, 2 VGPRs/lane): lanes 0-15 supply K=0,1; lanes 16-31 K=2,3.
    // With a = {local, 0}, row m of A holds {val[m], 0, val[m+16], 0}.
    // B = all-ones (layout-independent) -> D[m][*] = val[m] + val[m+16].
    // D layout: lanes 0-15 hold rows 0-7 in c[0..7]; lanes 16-31 hold rows 8-15.
    v2f a; a.x = local; a.y = 0.0f;
    v2f b; b.x = 1.0f;  b.y = 1.0f;
    v8f c = {};
    c = __builtin_amdgcn_wmma_f32_16x16x4_f32(
        /*neg_a=*/false, a, /*neg_b=*/false, b,
        /*c_mod=*/(short)0, c, /*reuse_a=*/false, /*reuse_b=*/false);
    float wsum = ((c[0] + c[1]) + (c[2] + c[3])) + ((c[4] + c[5]) + (c[6] + c[7]));
    wsum += __shfl_xor(wsum, 16, 32);   // combine lane halves -> full wave sum

    __shared__ float lds[BLOCK / 32];
    const int lane = threadIdx.x & 31;
    const int wave = threadIdx.x >> 5;
    if (lane == 0) lds[wave] = wsum;
    __syncthreads();

    if (threadIdx.x == 0) {
        float s = 0.0f;
#pragma unroll
        for (int i = 0; i < BLOCK / 32; ++i) s += lds[i];
        partials[blockIdx.x] = s;   // fixed slot, overwritten every call
    }
}

__global__ void __launch_bounds__(BLOCK)
pose_loss_final(const float* __restrict__ partials, int n, float* __restrict__ out) {
    __shared__ float lds[BLOCK];
    float s = 0.0f;
    for (int i = threadIdx.x; i < n; i += BLOCK) s += partials[i];  // fixed order
    lds[threadIdx.x] = s;
    __syncthreads();
#pragma unroll
    for (int off = BLOCK / 2; off > 0; off >>= 1) {
        if (threadIdx.x < off) lds[threadIdx.x] += lds[threadIdx.x + off];
        __syncthreads();
    }
    if (threadIdx.x == 0) out[0] = lds[0];
}

extern "C" void kernel_launch(void* const* d_in, const int* in_sizes, int n_in,
                              void* d_out, int out_size, void* d_ws, size_t ws_size,
                              hipStream_t stream) {
    const float* pred = (const float*)d_in[0];
    const float* tgt  = (const float*)d_in[1];

    const long B        = (long)in_sizes[0] / 7;          // 8388608
    const long nthreads = (B + ROWS_PER_THREAD - 1) / ROWS_PER_THREAD;
    const int  nblocks  = (int)((nthreads + BLOCK - 1) / BLOCK);  // 8192

    const float invB  = 1.0f / (float)B;
    const float invB3 = invB / 3.0f;

    float* partials = (float*)d_ws;   // nblocks floats (32 KB)

    pose_loss_main<<<nblocks, BLOCK, 0, stream>>>(pred, tgt, partials, B, invB3, invB);
    pose_loss_final<<<1, BLOCK, 0, stream>>>(partials, nblocks, (float*)d_out);
}